// Expression_Independent_AU_Loss_15582141350755
// MI455X (gfx1250) — compile-verified
//
#include <hip/hip_runtime.h>

// ---------------------------------------------------------------------------
// Expression-independent AU loss, MI455X (gfx1250).
// Memory-bound streaming reduction (256 MiB -> ~11us @ 23.3 TB/s).
// Gram matrix Ppos^T * Ppos computed with V_WMMA_F32_16X16X4_F32.
// Data path: global_load_async_to_lds_b128 (ASYNCcnt ring buffer) -> LDS ->
// WMMA-layout ds loads -> WMMA accumulate -> ds_add_f32 block reduce ->
// global_atomic_add_f32 -> tiny epilogue kernel.
// All tile-loop control flow is forced wave-uniform (readfirstlane) so the
// compiler emits scalar branches instead of EXEC-mask juggling.
// ---------------------------------------------------------------------------

typedef float v2f __attribute__((ext_vector_type(2)));
typedef float v8f __attribute__((ext_vector_type(8)));

#define TILE_ROWS 32
#define TILE_BYTES (TILE_ROWS * 16 * 4) /* 2048 B per tile */
#define DEPTH 3                         /* ring-buffer depth per wave */
#define WPB 8                           /* waves per block */
#define TPB 256

// 10 POS pairs then 4 NEG pairs
__constant__ int PAIR_I[14] = {0, 2, 2, 5, 4, 6, 9, 9, 11, 13, 1, 1, 8, 8};
__constant__ int PAIR_J[14] = {1, 5, 6, 6, 8, 11, 11, 14, 14, 14, 4, 5, 9, 11};

// Issue one 2KB tile copy (4 x b128 per lane) global -> LDS, tracked by ASYNCcnt.
// INST_OFFSET applies to both the global and the LDS address (ISA 08 §4.4).
__device__ __forceinline__ void issue_tile(unsigned ldsOff, unsigned gOff,
                                           unsigned long long base) {
  asm volatile(
      "s_wait_dscnt 0\n\t" // WAR: prior ds reads of this buffer must retire
      "global_load_async_to_lds_b128 %0, %1, %2\n\t"
      "global_load_async_to_lds_b128 %0, %1, %2 offset:512\n\t"
      "global_load_async_to_lds_b128 %0, %1, %2 offset:1024\n\t"
      "global_load_async_to_lds_b128 %0, %1, %2 offset:1536"
      :
      : "v"(ldsOff), "v"(gOff), "s"(base)
      : "memory");
}

__launch_bounds__(TPB) __global__
void au_main(const float* __restrict__ pred, float* __restrict__ ws, int nTiles) {
  __shared__ float tiles[WPB][DEPTH][TILE_ROWS * 16]; // 48 KB
  __shared__ float sPos[16];
  __shared__ float sNeg[16];
  __shared__ float sGram[256];

  const int tid = threadIdx.x;
  for (int i = tid; i < 288; i += TPB) {
    if (i < 16)      sPos[i] = 0.0f;
    else if (i < 32) sNeg[i - 16] = 0.0f;
    else             sGram[i - 32] = 0.0f;
  }
  __syncthreads();

  const int lane = tid & 31;
  const int wave = tid >> 5;

  // Wave-uniform loop state in SGPRs: scalar branches, EXEC stays all-ones.
  const int gwave  = __builtin_amdgcn_readfirstlane(blockIdx.x * WPB + wave);
  const int nWaves = (int)(gridDim.x) * WPB;

  const int c     = lane & 15; // column handled by this lane (A: M, B: N)
  const int khalf = lane >> 4; // 0 -> K={0,1}, 1 -> K={2,3} within a 4-row group
  const int rbase = khalf * 2;

  v8f acc0 = {0.f, 0.f, 0.f, 0.f, 0.f, 0.f, 0.f, 0.f};
  v8f acc1 = {0.f, 0.f, 0.f, 0.f, 0.f, 0.f, 0.f, 0.f};
  float possum = 0.0f, negsum = 0.0f;

  // LDS byte offset of this wave's ring buffer (+ per-lane 16B chunk offset).
  // Flat LDS addresses carry the LDS byte offset in the low 32 bits.
  const unsigned ldsLane =
      (unsigned)(unsigned long long)(const void*)&tiles[wave][0][0] +
      (unsigned)lane * 16u;
  const unsigned long long base = (unsigned long long)(const void*)pred;

  int inflight = 0;
  for (int d = 0; d < DEPTH; ++d) {
    int t = gwave + d * nWaves;
    if (t < nTiles) {
      issue_tile(ldsLane + (unsigned)d * TILE_BYTES,
                 (unsigned)t * TILE_BYTES + (unsigned)lane * 16u, base);
      ++inflight;
    }
  }

  int pb = 0;
  for (int t = gwave; t < nTiles; t += nWaves) {
    // Oldest copy done when at most 4*(inflight-1) async ops remain.
    if (inflight >= 3)      asm volatile("s_wait_asynccnt 8" ::: "memory");
    else if (inflight == 2) asm volatile("s_wait_asynccnt 4" ::: "memory");
    else                    asm volatile("s_wait_asynccnt 0" ::: "memory");

    const float* buf = &tiles[wave][0][0] + pb * (TILE_ROWS * 16);
#pragma unroll
    for (int g = 0; g < TILE_ROWS / 4; ++g) {
      const int r = g * 4 + rbase;
      float x0 = buf[r * 16 + c];
      float x1 = buf[(r + 1) * 16 + c];
      float p0 = (x0 >= 0.5f) ? x0 : 0.0f;
      float p1 = (x1 >= 0.5f) ? x1 : 0.0f;
      possum += p0 + p1;
      negsum += (x0 - p0) + (x1 - p1);
      v2f a;
      a.x = p0;
      a.y = p1;
      // D = A^T A accumulate: same registers feed SRC0 (A) and SRC1 (B).
      if (g & 1)
        acc1 = __builtin_amdgcn_wmma_f32_16x16x4_f32(false, a, false, a,
                                                     (short)0, acc1, false, false);
      else
        acc0 = __builtin_amdgcn_wmma_f32_16x16x4_f32(false, a, false, a,
                                                     (short)0, acc0, false, false);
    }

    const int tp = t + DEPTH * nWaves;
    if (tp < nTiles) {
      issue_tile(ldsLane + (unsigned)pb * TILE_BYTES,
                 (unsigned)tp * TILE_BYTES + (unsigned)lane * 16u, base);
    } else {
      --inflight;
    }
    pb = pb + 1;
    if (pb == DEPTH) pb = 0;
  }

  // Block-level reduction (ds_add_f32 shared atomics).
  atomicAdd(&sPos[c], possum);
  atomicAdd(&sNeg[c], negsum);
#pragma unroll
  for (int v = 0; v < 8; ++v) {
    // D layout: VGPR v holds (M = v + 8*lanehalf, N = lane%16).
    atomicAdd(&sGram[(v + 8 * khalf) * 16 + c], acc0[v] + acc1[v]);
  }
  __syncthreads();

  // Flush only the 46 needed scalars to global workspace.
  for (int i = tid; i < 46; i += TPB) {
    float v;
    if (i < 16)      v = sPos[i];
    else if (i < 32) v = sNeg[i - 16];
    else {
      int k = i - 32;
      v = sGram[PAIR_I[k] * 16 + PAIR_J[k]];
    }
    atomicAdd(&ws[i], v);
  }
}

__global__ void au_zero(float* __restrict__ ws) {
  if (threadIdx.x < 46) ws[threadIdx.x] = 0.0f;
}

// Remainder rows + 14-pair epilogue (negligible work, single thread).
__global__ void au_final(const float* __restrict__ pred,
                         const float* __restrict__ ws, float* __restrict__ out,
                         int rows, int tailStart) {
  if (threadIdx.x != 0 || blockIdx.x != 0) return;
  float pos[16], neg[16], pp[14];
  for (int i = 0; i < 16; ++i) { pos[i] = ws[i]; neg[i] = ws[16 + i]; }
  for (int k = 0; k < 14; ++k) pp[k] = ws[32 + k];

  for (int r = tailStart; r < rows; ++r) {
    float x[16];
    for (int cc = 0; cc < 16; ++cc) {
      float v = pred[r * 16 + cc];
      x[cc] = v;
      if (v >= 0.5f) pos[cc] += v; else neg[cc] += v;
    }
    for (int k = 0; k < 14; ++k) {
      float a = x[PAIR_I[k]], b = x[PAIR_J[k]];
      if (a >= 0.5f && b >= 0.5f) pp[k] += a * b;
    }
  }

  const float inv_n = 1.0f / (float)rows;
  float loss = 0.0f;
  for (int k = 0; k < 10; ++k) { // POS pairs
    int i = PAIR_I[k], j = PAIR_J[k];
    float Pi = pos[i] * inv_n, Pj = pos[j] * inv_n;
    float Ni = neg[i] * inv_n, Nj = neg[j] * inv_n;
    float PP = pp[k] * inv_n;
    loss += fmaxf(Pi * Pj - PP, 0.0f) + fmaxf(Ni * Pj - PP, 0.0f) +
            fmaxf(Pi * Nj - PP, 0.0f);
  }
  for (int k = 10; k < 14; ++k) { // NEG pairs
    int i = PAIR_I[k], j = PAIR_J[k];
    float Pi = pos[i] * inv_n, Pj = pos[j] * inv_n;
    float Ni = neg[i] * inv_n, Nj = neg[j] * inv_n;
    float PP = pp[k] * inv_n;
    loss += fmaxf(Pi * Pj - PP, 0.0f) + fmaxf(PP - Ni * Pj, 0.0f) +
            fmaxf(PP - Pi * Nj, 0.0f);
  }
  out[0] = loss;
}

extern "C" void kernel_launch(void* const* d_in, const int* in_sizes, int n_in,
                              void* d_out, int out_size, void* d_ws,
                              size_t ws_size, hipStream_t stream) {
  const float* pred = (const float*)d_in[0];
  float* out = (float*)d_out;
  float* ws = (float*)d_ws;

  const long long total = (long long)in_sizes[0];
  const int rows = (int)(total / 16);
  const int nTiles = rows / TILE_ROWS;

  au_zero<<<1, 64, 0, stream>>>(ws);

  int blocks = 512;
  long long maxB = ((long long)nTiles + WPB - 1) / WPB;
  if (maxB < 1) maxB = 1;
  if (maxB < blocks) blocks = (int)maxB;

  au_main<<<blocks, TPB, 0, stream>>>(pred, ws, nTiles);
  au_final<<<1, 32, 0, stream>>>(pred, ws, out, rows, nTiles * TILE_ROWS);
}